// LIF_22093311770853
// MI455X (gfx1250) — compile-verified
//
#include <hip/hip_runtime.h>
#include <stdint.h>

// LIF forward scan: u = 0.5u + x - 0.5*o ; o = (u > 0.5)
// 262,144 independent sequences x 200 steps, fp32. Pure bandwidth problem
// (420 MB traffic, ~18us floor @ 23.3 TB/s). One thread per sequence;
// input staged through LDS via CDNA5 async-to-LDS copies (ASYNCcnt),
// double-buffered across time chunks. Both streams are touch-once, so all
// global traffic uses non-temporal cache policy (don't pollute 192MB L2).

#define STEPS      200
#define TC         40            // time steps per chunk (160B per row: 16B aligned)
#define NCHUNK     (STEPS / TC)  // 5
#define ROWS       128           // sequences per block == blockDim.x (4 wave32 waves)
#define LDS_STRIDE 44            // padded row stride in floats (176B: 16B aligned, 2-way DS conflict)
#define LOADS_PER_THREAD 10      // (ROWS rows * 10 float4-chunks/row) / ROWS threads
#define DECAY 0.5f
#define VTH   0.5f

typedef __attribute__((ext_vector_type(4))) float f4;

__device__ __forceinline__ void async_load_f4(uint32_t lds_byte_off, const float* gaddr) {
    // CDNA5 GLOBAL_LOAD_ASYNC_TO_LDS_B128 (GV mode): vdst = LDS byte address,
    // vaddr = 64-bit global address. Tracked by ASYNCcnt, completes in order.
    // Touch-once stream -> non-temporal hint.
    asm volatile("global_load_async_to_lds_b128 %0, %1, off th:TH_LOAD_NT"
                 :
                 : "v"(lds_byte_off), "v"(gaddr)
                 : "memory");
}

__global__ __launch_bounds__(ROWS) void lif_scan_kernel(const float* __restrict__ x,
                                                        float* __restrict__ out) {
    __shared__ __align__(16) float smem[2][ROWS * LDS_STRIDE];

    const int tid = threadIdx.x;
    const long long row0 = (long long)blockIdx.x * ROWS;
    const float* xb = x + row0 * STEPS;
    float*       ob = out + row0 * STEPS;

    const uint32_t lbase0 = (uint32_t)(uintptr_t)(&smem[0][0]);
    const uint32_t lbase1 = (uint32_t)(uintptr_t)(&smem[1][0]);

    // Issue one chunk's worth of async tile loads (10 B128 per thread).
    // Lanes walk the tile row-major (j = i*ROWS + tid), so each wave
    // instruction covers a dense run of memory (~512B across <=4 row tails).
    auto issue = [&](int c, uint32_t lbase) {
        const float* gbase = xb + c * TC;
#pragma unroll
        for (int i = 0; i < LOADS_PER_THREAD; ++i) {
            int j = i * ROWS + tid;           // 0 .. 1279
            int r = j / LOADS_PER_THREAD;     // row in tile
            int k = j - r * LOADS_PER_THREAD; // float4-chunk within row
            async_load_f4(lbase + (uint32_t)(r * LDS_STRIDE + k * 4) * 4u,
                          gbase + (size_t)r * STEPS + (size_t)k * 4);
        }
    };

    issue(0, lbase0);

    float u = 0.0f, o = 0.0f;

#pragma unroll
    for (int c = 0; c < NCHUNK; ++c) {
        const int p = c & 1;
        if (c + 1 < NCHUNK) {
            issue(c + 1, (p ^ 1) ? lbase1 : lbase0);
            // Loads retire in order per wave: cnt<=10 => chunk c's 10 loads landed,
            // only the 10 just issued for chunk c+1 may remain outstanding.
            asm volatile("s_wait_asynccnt 10" ::: "memory");
        } else {
            asm volatile("s_wait_asynccnt 0" ::: "memory");
        }
        __syncthreads();  // every wave has drained its share of the tile into LDS

        const f4* row  = (const f4*)(&smem[p][tid * LDS_STRIDE]);
        float*    orow = ob + (size_t)tid * STEPS + (size_t)c * TC;
#pragma unroll
        for (int tt = 0; tt < TC / 4; ++tt) {
            f4 xi = row[tt];  // ds_load_b128
            f4 ov;
            u = DECAY * u + xi.x - o * VTH; o = (u > VTH) ? 1.0f : 0.0f; ov.x = o;
            u = DECAY * u + xi.y - o * VTH; o = (u > VTH) ? 1.0f : 0.0f; ov.y = o;
            u = DECAY * u + xi.z - o * VTH; o = (u > VTH) ? 1.0f : 0.0f; ov.z = o;
            u = DECAY * u + xi.w - o * VTH; o = (u > VTH) ? 1.0f : 0.0f; ov.w = o;
            // contiguous per-lane 160B/chunk; write-once stream -> NT store
            __builtin_nontemporal_store(ov, (f4*)(orow + tt * 4));
        }
        __syncthreads();  // all waves done reading buf[p] before it is refilled
    }
}

extern "C" void kernel_launch(void* const* d_in, const int* in_sizes, int n_in,
                              void* d_out, int out_size, void* d_ws, size_t ws_size,
                              hipStream_t stream) {
    (void)n_in; (void)out_size; (void)d_ws; (void)ws_size;
    const float* x = (const float*)d_in[0];
    float* out = (float*)d_out;
    const int n_rows = in_sizes[0] / STEPS;   // 32*8192 = 262144
    const int blocks = n_rows / ROWS;         // 2048
    lif_scan_kernel<<<dim3(blocks), dim3(ROWS), 0, stream>>>(x, out);
}